// Self_Attention_14044543058310
// MI455X (gfx1250) — compile-verified
//
#include <hip/hip_runtime.h>
#include <hip/hip_bf16.h>

typedef float v2f __attribute__((ext_vector_type(2)));
typedef float v4f __attribute__((ext_vector_type(4)));
typedef float v8f __attribute__((ext_vector_type(8)));
typedef unsigned int u32x4 __attribute__((ext_vector_type(4)));
typedef int i32x4 __attribute__((ext_vector_type(4)));
typedef int i32x8 __attribute__((ext_vector_type(8)));

#define BB 4
#define LL 4096
#define DD 512
#define SCALE_INV 0.0625f /* 1/sqrt(HIDDEN_DIM=256) */

#if defined(__gfx1250__) && __has_builtin(__builtin_amdgcn_tensor_load_to_lds)
#define USE_TDM 1
#else
#define USE_TDM 0
#endif

__device__ __forceinline__ v8f wmma_f32(v2f a, v2f b, v8f c) {
    // D = A(16x4 f32) * B(4x16 f32) + C(16x16 f32)
    return __builtin_amdgcn_wmma_f32_16x16x4_f32(
        /*neg_a=*/false, a, /*neg_b=*/false, b,
        /*c_mod=*/(short)0, c, /*reuse_a=*/false, /*reuse_b=*/false);
}

#if USE_TDM
// Issue one TDM 2-D tile load: `rows` rows x 512 fp32 columns, row stride 512
// elements, from gptr into LDS at lds_byte_addr. Tracked by TENSORcnt.
// D# packing per CDNA5 ISA §8.3/8.4 (group0 128b, group1 256b; groups 2/3
// unused for 2-D tensors -> zeros). 6-arg builtin form (clang-23 lane).
__device__ __forceinline__ void tdm_load_rows_512(const float* gptr,
                                                  unsigned int lds_byte_addr,
                                                  int rows) {
    unsigned long long ga = (unsigned long long)(uintptr_t)gptr;
    u32x4 g0;
    g0[0] = 1u;                                        // count=1 (valid), user mode
    g0[1] = lds_byte_addr;                             // lds_addr[31:0]
    g0[2] = (unsigned int)(ga & 0xFFFFFFFFu);          // global_addr[31:0]
    g0[3] = (unsigned int)((ga >> 32) & 0x1FFFFFFu)    // global_addr[56:32]
            | (2u << 30);                              // type=2 ("image")
    i32x8 g1;
    g1[0] = (int)(2u << 16);                           // wg_mask=0, data_size=2 (4B)
    g1[1] = (int)(512u << 16);                         // tensor_dim0=512 (lo16)
    g1[2] = (int)(16384u << 16);                       // dim0 hi=0 | tensor_dim1=16384 (lo16)
    g1[3] = (int)(512u << 16);                         // dim1 hi=0 | tile_dim0=512
    g1[4] = rows & 0xFFFF;                             // tile_dim1=rows, tile_dim2=0
    g1[5] = 512;                                       // tensor_dim0_stride lo32
    g1[6] = 0;                                         // stride hi | dim1_stride lo
    g1[7] = 0;                                         // dim1_stride hi
    i32x4 z4 = {0, 0, 0, 0};
    i32x8 z8 = {0, 0, 0, 0, 0, 0, 0, 0};
    __builtin_amdgcn_tensor_load_to_lds(g0, g1, z4, z4, z8, 0);
}
#endif

__global__ void zero_kernel(float* __restrict__ p, int n) {
    int i = blockIdx.x * blockDim.x + threadIdx.x;
    if (i < n) p[i] = 0.0f;
}

// ---------------------------------------------------------------------------
// Kernel 1: Q/K/V projection.  O[r, j] = sum_d X[r, d] * W[j, d]
// grid.x = (B*L)/16 row tiles, grid.z = {0:Q, 1:K, 2:V}; 256 thr = 8 waves.
// TDM stages the 16x512 X row-tile into LDS; wave w computes cols [w*64, +64).
// ---------------------------------------------------------------------------
__global__ __launch_bounds__(256, 2)
void qkv_kernel(const float* __restrict__ X,
                const float* __restrict__ Wq, const float* __restrict__ Wk,
                const float* __restrict__ Wv,
                float* __restrict__ Q, float* __restrict__ K, float* __restrict__ V) {
    __shared__ float ldsA[16 * DD];                     // 32 KB
    const float* W = (blockIdx.z == 0) ? Wq : ((blockIdx.z == 1) ? Wk : Wv);
    float*       O = (blockIdx.z == 0) ? Q  : ((blockIdx.z == 1) ? K  : V);

    const int t  = threadIdx.x;
    const int rt = blockIdx.x;                          // 16-row tile index
    const float* src = X + (size_t)rt * 16 * DD;
#if USE_TDM
    if (t < 32) {                                       // wave 0 drives the TDM
        tdm_load_rows_512(src, (unsigned int)(uintptr_t)&ldsA[0], 16);
        __builtin_amdgcn_s_wait_tensorcnt(0);
    }
#else
#pragma unroll
    for (int i = 0; i < 8; ++i) {                       // 16*512 floats / 256 thr
        int idx = (i * 256 + t) * 4;
        *(v4f*)&ldsA[idx] = *(const v4f*)&src[idx];
    }
#endif
    __syncthreads();

    const int wave = t >> 5;
    const int lane = t & 31;
    const int lr   = lane & 15;                         // row within fragment
    const int kh   = (lane >> 4) << 1;                  // k sub-offset 0 or 2
    const int ct   = wave * 64;

    v8f acc0 = {}, acc1 = {}, acc2 = {}, acc3 = {};
    const float* ap = &ldsA[lr * DD + kh];
    const float* w0 = W + (size_t)(ct +  0 + lr) * DD + kh;
    const float* w1 = W + (size_t)(ct + 16 + lr) * DD + kh;
    const float* w2 = W + (size_t)(ct + 32 + lr) * DD + kh;
    const float* w3 = W + (size_t)(ct + 48 + lr) * DD + kh;

#pragma unroll 4
    for (int kk = 0; kk < DD; kk += 4) {
        v2f a = *(const v2f*)(ap + kk);
        acc0 = wmma_f32(a, *(const v2f*)(w0 + kk), acc0);
        acc1 = wmma_f32(a, *(const v2f*)(w1 + kk), acc1);
        acc2 = wmma_f32(a, *(const v2f*)(w2 + kk), acc2);
        acc3 = wmma_f32(a, *(const v2f*)(w3 + kk), acc3);
    }

    const int mbase = (lane >> 4) ? 8 : 0;              // C/D layout: vgpr i -> M=i or 8+i
    float* orow = O + (size_t)(rt * 16) * DD + ct + lr;
#pragma unroll
    for (int i = 0; i < 8; ++i) {
        orow[(size_t)(mbase + i) * DD +  0] = acc0[i];
        orow[(size_t)(mbase + i) * DD + 16] = acc1[i];
        orow[(size_t)(mbase + i) * DD + 32] = acc2[i];
        orow[(size_t)(mbase + i) * DD + 48] = acc3[i];
    }
}

// ---------------------------------------------------------------------------
// Kernel 2: fused scores + batch-axis softmax + query mask + column-sum.
// grid.x = 256 key tiles, grid.y = 8 query groups (32 q-tiles each).
// TDM stages the K-tile for all 4 batches into dynamic LDS (128 KB).
// Wave w sweeps q-tiles {gy*32 + w + 8*qi}; per q-tile: 4 batch accumulators,
// 128 f32-WMMA steps each, then softmax across b, mask, colsum accumulate.
// ---------------------------------------------------------------------------
__global__ __launch_bounds__(256, 1)
void score_kernel(const float* __restrict__ Q, const float* __restrict__ K,
                  const int* __restrict__ lens, float* __restrict__ colsum) {
    extern __shared__ float smem[];
    float* ldsK = smem;                                  // [b][keyrow 16][512]
    float* red  = smem + BB * 16 * DD;                   // [wave 8][b 4][n 16]

    const int t  = threadIdx.x;
    const int kt = blockIdx.x;                           // key tile (per-batch l/16)

#if USE_TDM
    if (t < 32) {                                        // wave 0 drives 4 TDM tiles
        unsigned int lbase = (unsigned int)(uintptr_t)&ldsK[0];
#pragma unroll
        for (int b = 0; b < BB; ++b)
            tdm_load_rows_512(K + ((size_t)b * LL + kt * 16) * DD,
                              lbase + (unsigned int)(b * 16 * DD * 4), 16);
        __builtin_amdgcn_s_wait_tensorcnt(0);
    }
#else
#pragma unroll
    for (int i = 0; i < 32; ++i) {                       // 4*16*512 floats / 256 thr
        int idx = (i * 256 + t) * 4;
        int row = idx >> 9;
        int c   = idx & 511;
        int b   = row >> 4, r = row & 15;
        *(v4f*)&ldsK[idx] =
            *(const v4f*)&K[((size_t)b * LL + kt * 16 + r) * DD + c];
    }
#endif
    __syncthreads();

    const int wave  = t >> 5, lane = t & 31;
    const int lr    = lane & 15;
    const int kh    = (lane >> 4) << 1;
    const int mbase = (lane >> 4) ? 8 : 0;

    const int len0 = lens[0], len1 = lens[1], len2 = lens[2], len3 = lens[3];
    float cs0 = 0.f, cs1 = 0.f, cs2 = 0.f, cs3 = 0.f;

    const float* kb0 = &ldsK[( 0 + lr) * DD + kh];
    const float* kb1 = &ldsK[(16 + lr) * DD + kh];
    const float* kb2 = &ldsK[(32 + lr) * DD + kh];
    const float* kb3 = &ldsK[(48 + lr) * DD + kh];

    for (int qi = 0; qi < 4; ++qi) {
        const int qt = blockIdx.y * 32 + qi * 8 + wave;  // 0..255
        v8f s0 = {}, s1 = {}, s2 = {}, s3 = {};
        const float* q0 = Q + ((size_t)0 * LL + qt * 16 + lr) * DD + kh;
        const float* q1 = Q + ((size_t)1 * LL + qt * 16 + lr) * DD + kh;
        const float* q2 = Q + ((size_t)2 * LL + qt * 16 + lr) * DD + kh;
        const float* q3 = Q + ((size_t)3 * LL + qt * 16 + lr) * DD + kh;

#pragma unroll 2
        for (int kk = 0; kk < DD; kk += 4) {
            s0 = wmma_f32(*(const v2f*)(q0 + kk), *(const v2f*)(kb0 + kk), s0);
            s1 = wmma_f32(*(const v2f*)(q1 + kk), *(const v2f*)(kb1 + kk), s1);
            s2 = wmma_f32(*(const v2f*)(q2 + kk), *(const v2f*)(kb2 + kk), s2);
            s3 = wmma_f32(*(const v2f*)(q3 + kk), *(const v2f*)(kb3 + kk), s3);
        }

        // 4-way softmax across batch per (q,k) element, then query-row mask.
#pragma unroll
        for (int i = 0; i < 8; ++i) {
            float a0 = s0[i] * SCALE_INV, a1 = s1[i] * SCALE_INV;
            float a2 = s2[i] * SCALE_INV, a3 = s3[i] * SCALE_INV;
            float m  = fmaxf(fmaxf(a0, a1), fmaxf(a2, a3));
            float e0 = __expf(a0 - m), e1 = __expf(a1 - m);
            float e2 = __expf(a2 - m), e3 = __expf(a3 - m);
            float inv = 1.0f / (e0 + e1 + e2 + e3);
            int qrow = qt * 16 + mbase + i;
            cs0 += (qrow < len0) ? e0 * inv : 0.f;
            cs1 += (qrow < len1) ? e1 * inv : 0.f;
            cs2 += (qrow < len2) ? e2 * inv : 0.f;
            cs3 += (qrow < len3) ? e3 * inv : 0.f;
        }
    }

    // lanes n and n+16 hold the same key column -> fold halves (wave32).
    cs0 += __shfl_xor(cs0, 16, 32);
    cs1 += __shfl_xor(cs1, 16, 32);
    cs2 += __shfl_xor(cs2, 16, 32);
    cs3 += __shfl_xor(cs3, 16, 32);
    if (lane < 16) {
        red[(wave * 4 + 0) * 16 + lane] = cs0;
        red[(wave * 4 + 1) * 16 + lane] = cs1;
        red[(wave * 4 + 2) * 16 + lane] = cs2;
        red[(wave * 4 + 3) * 16 + lane] = cs3;
    }
    __syncthreads();
    if (t < 64) {
        int b = t >> 4, n = t & 15;
        float s = 0.f;
#pragma unroll
        for (int w = 0; w < 8; ++w) s += red[(w * 4 + b) * 16 + n];
        atomicAdd(&colsum[b * LL + kt * 16 + n], s);
    }
}

// ---------------------------------------------------------------------------
// Kernel 3: out[b, d] = sum_k colsum[b, k] * V[b, k, d]   (tiny GEMV)
// ---------------------------------------------------------------------------
__global__ void out_kernel(const float* __restrict__ colsum,
                           const float* __restrict__ V, float* __restrict__ out) {
    int idx = blockIdx.x * blockDim.x + threadIdx.x;     // 0..2047
    int b = idx >> 9, d = idx & 511;
    const float* cs = colsum + b * LL;
    const float* v  = V + (size_t)b * LL * DD + d;
    float acc = 0.f;
#pragma unroll 4
    for (int k = 0; k < LL; ++k) acc += cs[k] * v[(size_t)k * DD];
    out[idx] = acc;
}

extern "C" void kernel_launch(void* const* d_in, const int* in_sizes, int n_in,
                              void* d_out, int out_size, void* d_ws, size_t ws_size,
                              hipStream_t stream) {
    const float* X    = (const float*)d_in[0];
    const float* Wq   = (const float*)d_in[1];
    const float* Wk   = (const float*)d_in[2];
    const float* Wv   = (const float*)d_in[3];
    const int*   lens = (const int*)d_in[4];
    float*       out  = (float*)d_out;

    const size_t mat = (size_t)BB * LL * DD;             // 8,388,608 floats
    float* Q      = (float*)d_ws;
    float* K      = Q + mat;
    float* V      = K + mat;
    float* colsum = V + mat;                             // B*L floats

    zero_kernel<<<(BB * LL + 255) / 256, 256, 0, stream>>>(colsum, BB * LL);

    dim3 gQKV((BB * LL) / 16, 1, 3);
    qkv_kernel<<<gQKV, 256, 0, stream>>>(X, Wq, Wk, Wv, Q, K, V);

    dim3 gS(LL / 16, 8, 1);
    size_t smem = (size_t)(BB * 16 * DD + 8 * 4 * 16) * sizeof(float); // ~130 KB
    score_kernel<<<gS, 256, smem, stream>>>(Q, K, lens, colsum);

    out_kernel<<<(BB * DD + 255) / 256, 256, 0, stream>>>(colsum, V, out);
}